// Encoder_64613488001206
// MI455X (gfx1250) — compile-verified
//
#include <hip/hip_runtime.h>
#include <stdint.h>

// ---------------- problem constants (from reference) ----------------
#define UNITF      0.1f
#define HALF_UNITF 0.05f            // UNIT/2 added in voxel units (as in source)
#define KEY_OFF    (1ll << 20)
#define KEY_BASE   (1ll << 21)
#define CCH        128              // feature channels
#define TILE_PTS   64               // points staged per block in accumulate

// ---------------- hash table ----------------
#define CAP        (1u << 19)       // 2x max unique voxels -> load factor <= 0.5
#define CAPMASK    (CAP - 1u)
#define EMPTY_KEY  0xFFFFFFFFFFFFFFFFull

typedef unsigned int v4u __attribute__((ext_vector_type(4)));
typedef int          v8i __attribute__((ext_vector_type(8)));
typedef int          v4i __attribute__((ext_vector_type(4)));
typedef float        fv4 __attribute__((ext_vector_type(4)));

__device__ __forceinline__ uint32_t hash_slot(uint64_t k) {
    k *= 0x9E3779B97F4A7C15ull;
    k ^= k >> 32;
    return (uint32_t)k & CAPMASK;
}

// ---------------- K2: TDM-stage feature tile to LDS, scatter-add to sums ----------------
// (placed first in the file so the disasm snippet shows the TDM codegen)
__global__ void __launch_bounds__(256)
k_accum(const float* __restrict__ feats, const uint32_t* __restrict__ slots,
        float* __restrict__ sums) {
    __shared__ float tile[TILE_PTS * CCH];         // 64 pts x 128 ch x f32 = 32 KB
    const int base = blockIdx.x * TILE_PTS;

#if defined(__has_builtin) && __has_builtin(__builtin_amdgcn_tensor_load_to_lds)
    if ((threadIdx.x >> 5) == 0) {                 // wave 0 issues the DMA (EXEC ignored by TDM)
        uint64_t gaddr = (uint64_t)(uintptr_t)(feats + (size_t)base * CCH);
        uint32_t laddr = (uint32_t)(uintptr_t)&tile[0];
        // D# group 0: count=1 | lds_addr | global_addr[56:0] | type=2
        v4u g0;
        g0[0] = 1u;
        g0[1] = laddr;
        g0[2] = (uint32_t)gaddr;
        g0[3] = (uint32_t)((gaddr >> 32) & 0x01FFFFFFu) | (2u << 30);
        // D# group 1: data_size=4B(code 2); tensor_dim0=128; tensor_dim1=262144;
        //             tile_dim0=128, tile_dim1=64; tensor_dim0_stride=128
        v8i g1;
        g1[0] = (int)(2u << 16);                               // data_size
        g1[1] = (int)(((uint32_t)CCH & 0xFFFFu) << 16);        // tensor_dim0 lo16
        g1[2] = (int)((0x40000u & 0xFFFFu) << 16);             // dim0 hi16=0 | dim1 lo16
        g1[3] = (int)((0x40000u >> 16) | (((uint32_t)CCH) << 16)); // dim1 hi16 | tile_dim0
        g1[4] = (int)TILE_PTS;                                 // tile_dim1 | tile_dim2=0
        g1[5] = (int)CCH;                                      // tensor_dim0_stride lo32
        g1[6] = 0;
        g1[7] = 0;
        v4i gz4 = {0, 0, 0, 0};
        v8i gz8 = {0, 0, 0, 0, 0, 0, 0, 0};
        __builtin_amdgcn_tensor_load_to_lds(g0, g1, gz4, gz4, gz8, 0);
        __builtin_amdgcn_s_wait_tensorcnt(0);
    }
#else
    for (int t = threadIdx.x; t < TILE_PTS * CCH / 4; t += blockDim.x)
        ((fv4*)tile)[t] = ((const fv4*)(feats + (size_t)base * CCH))[t];
#endif
    __syncthreads();

    const int lane = threadIdx.x & 31;             // wave32: 32 lanes x float4 = 128 ch
    const int wave = threadIdx.x >> 5;
    for (int p = wave; p < TILE_PTS; p += 8) {
        uint32_t slot = slots[base + p];
        const fv4 v = *(const fv4*)&tile[p * CCH + lane * 4];
        float* dst = &sums[(size_t)slot * CCH + lane * 4];
        atomicAdd(dst + 0, v.x);
        atomicAdd(dst + 1, v.y);
        atomicAdd(dst + 2, v.z);
        atomicAdd(dst + 3, v.w);
    }
}

// ---------------- K0: init hash table + zero accumulators ----------------
__global__ void k_init(fv4* __restrict__ sums4,
                       unsigned long long* __restrict__ keys,
                       uint32_t* __restrict__ counts, int nsum4) {
    int i = blockIdx.x * blockDim.x + threadIdx.x;
    int stride = gridDim.x * blockDim.x;
    fv4 z = {0.f, 0.f, 0.f, 0.f};
    for (int t = i; t < nsum4; t += stride) sums4[t] = z;
    for (int t = i; t < (int)CAP; t += stride) { keys[t] = EMPTY_KEY; counts[t] = 0u; }
}

// ---------------- K1: voxel key -> hash insert (CAS linear probe) ----------------
__global__ void k_insert(const float* __restrict__ pts,
                         unsigned long long* __restrict__ keys,
                         uint32_t* __restrict__ counts,
                         uint32_t* __restrict__ slots, int n) {
    int i = blockIdx.x * blockDim.x + threadIdx.x;
    if (i >= n) return;
    if (i + 8192 < n) __builtin_prefetch(pts + (size_t)(i + 8192) * 3, 0, 0);

    float px = pts[(size_t)i * 3 + 0] / UNITF;
    float py = pts[(size_t)i * 3 + 1] / UNITF;
    float pz = pts[(size_t)i * 3 + 2] / UNITF;
    long long cx = (long long)truncf(px) + KEY_OFF;
    long long cy = (long long)truncf(py) + KEY_OFF;
    long long cz = (long long)truncf(pz) + KEY_OFF;
    unsigned long long key =
        (unsigned long long)(((cx * KEY_BASE) + cy) * KEY_BASE + cz);

    uint32_t idx = hash_slot(key);
    while (true) {
        unsigned long long prev = atomicCAS(&keys[idx], EMPTY_KEY, key);
        if (prev == EMPTY_KEY || prev == key) break;
        idx = (idx + 1u) & CAPMASK;
    }
    slots[i] = idx;
    atomicAdd(&counts[idx], 1u);
}

// ---------------- K3: gather mean, distance-weight, residual, concat ----------------
__global__ void __launch_bounds__(256)
k_final(const float* __restrict__ pts, const float* __restrict__ feats,
        const uint32_t* __restrict__ slots, const uint32_t* __restrict__ counts,
        const float* __restrict__ sums, float* __restrict__ out, int n) {
    int gid  = blockIdx.x * blockDim.x + threadIdx.x;
    int i    = gid >> 5;                           // one wave per point
    int lane = gid & 31;
    if (i >= n) return;

    uint32_t slot = slots[i];
    float cnt = fmaxf((float)counts[slot], 1.0f);

    float px = pts[(size_t)i * 3 + 0] / UNITF;
    float py = pts[(size_t)i * 3 + 1] / UNITF;
    float pz = pts[(size_t)i * 3 + 2] / UNITF;
    float dx = px - (truncf(px) + HALF_UNITF);
    float dy = py - (truncf(py) + HALF_UNITF);
    float dz = pz - (truncf(pz) + HALF_UNITF);
    float w = sqrtf(dx * dx + dy * dy + dz * dz) / cnt;  // dist * (1/count)

    fv4 f = ((const fv4*)(feats + (size_t)i * CCH))[lane];
    fv4 s = ((const fv4*)(sums + (size_t)slot * CCH))[lane];
    fv4 a;
    a.x = s.x * w + f.x;
    a.y = s.y * w + f.y;
    a.z = s.z * w + f.z;
    a.w = s.w * w + f.w;

    // streaming output: non-temporal stores, keep L2 for feats/sums residency
    fv4* o = (fv4*)(out + (size_t)i * 2 * CCH);
    __builtin_nontemporal_store(f, o + lane);        // concat[:, :, 0:128]
    __builtin_nontemporal_store(a, o + 32 + lane);   // concat[:, :, 128:256]
}

// ---------------- launch ----------------
extern "C" void kernel_launch(void* const* d_in, const int* in_sizes, int n_in,
                              void* d_out, int out_size, void* d_ws, size_t ws_size,
                              hipStream_t stream) {
    const float* pts   = (const float*)d_in[0];    // [B, N, 3]  f32
    const float* feats = (const float*)d_in[1];    // [B, N, C]  f32
    float*       out   = (float*)d_out;            // [B, N, 2C] f32

    const int B = 2;
    const int N = in_sizes[0] / (B * 3);           // 262144

    // workspace layout (reused per batch; kernels serialize on stream)
    char* ws = (char*)d_ws;
    float*              sums   = (float*)ws;                                   // CAP*128*4 = 256 MB
    unsigned long long* keys   = (unsigned long long*)(ws + (size_t)CAP * CCH * 4); // 4 MB
    uint32_t*           counts = (uint32_t*)((char*)keys + (size_t)CAP * 8);   // 2 MB
    uint32_t*           slots  = (uint32_t*)((char*)counts + (size_t)CAP * 4); // 1 MB

    const int nsum4 = (int)(CAP * (CCH / 4));

    for (int b = 0; b < B; ++b) {
        const float* pb = pts   + (size_t)b * N * 3;
        const float* fb = feats + (size_t)b * N * CCH;
        float*       ob = out   + (size_t)b * N * 2 * CCH;

        k_init  <<<4096, 256, 0, stream>>>((fv4*)sums, keys, counts, nsum4);
        k_insert<<<(N + 255) / 256, 256, 0, stream>>>(pb, keys, counts, slots, N);
        k_accum <<<N / TILE_PTS, 256, 0, stream>>>(fb, slots, sums);
        k_final <<<(N * 32 + 255) / 256, 256, 0, stream>>>(pb, fb, slots, counts, sums, ob, N);
    }
}